// GCN_1949915152819
// MI455X (gfx1250) — compile-verified
//
#include <hip/hip_runtime.h>
#include <math.h>

typedef __attribute__((ext_vector_type(2))) float v2f;
typedef __attribute__((ext_vector_type(4))) float v4f;
typedef __attribute__((ext_vector_type(8))) float v8f;

#define DIM_IN  512
#define DIM_H   128
#define DIM_OUT 64

// ---------------------------------------------------------------------------
// Degree / normalization
// ---------------------------------------------------------------------------
__global__ void deg_init_kernel(float* __restrict__ deg, int n) {
  int i = blockIdx.x * blockDim.x + threadIdx.x;
  if (i < n) deg[i] = 1.0f;  // self-loop contributes 1
}

__global__ void deg_accum_kernel(const long long* __restrict__ dst,
                                 float* __restrict__ deg, int E) {
  int stride = gridDim.x * blockDim.x;
  for (int e = blockIdx.x * blockDim.x + threadIdx.x; e < E; e += stride)
    atomicAdd(&deg[(int)dst[e]], 1.0f);
}

__global__ void deg_rsqrt_kernel(float* __restrict__ deg, int n) {
  int i = blockIdx.x * blockDim.x + threadIdx.x;
  if (i < n) deg[i] = rsqrtf(deg[i]);
}

// ---------------------------------------------------------------------------
// fp32 WMMA GEMM: C[M x BN] = A[M x K] * B[K x BN], BN == full output width.
// Block tile 64 x BN; 8 waves (4 row x 2 col); wave tile 16 x (NT*16).
// A panel streamed with non-temporal loads (read exactly once per GEMM);
// B panel (weights) default-temporal: reused by every block via L2.
// B staged pair-interleaved in LDS so each lane's (B[k][n], B[k+1][n]) is one
// contiguous 8B read -> ds_load_b64, no packing movs in the WMMA loop.
// ---------------------------------------------------------------------------
template <int K, int BN, int NT>
__global__ __launch_bounds__(256) void gemm_wmma_f32(const float* __restrict__ A,
                                                     const float* __restrict__ B,
                                                     float* __restrict__ C, int M) {
  constexpr int TK  = 32;
  constexpr int AP  = TK + 2;   // A row pad: 34-dword stride -> conflict-free b64
  constexpr int PBN = BN + 16;  // pair-row stride*2 = 32 mod 64 banks

  __shared__ float As[64][AP];
  __shared__ float BsP[TK / 2][PBN][2];

  const int t    = threadIdx.x;
  const int wave = t >> 5;
  const int lane = t & 31;
  const int wrow = wave >> 1;        // 0..3
  const int wcol = wave & 1;         // 0..1
  const int m0w  = wrow * 16;        // wave row offset within block tile
  const int n0w  = wcol * (NT * 16); // wave col offset
  const int half = lane >> 4;        // lane half selects K pair / row half
  const int l    = lane & 15;

  const int rowBase = blockIdx.x * 64;

  v8f z = {};
  v8f acc[NT];
#pragma unroll
  for (int i = 0; i < NT; ++i) acc[i] = z;

  for (int kc = 0; kc < K; kc += TK) {
    // Stage A chunk: 64 x 32 floats = 512 v4f, 2 per thread (non-temporal)
#pragma unroll
    for (int i = 0; i < 2; ++i) {
      int idx = t + i * 256;     // 0..511
      int r   = idx >> 3;        // / (TK/4)
      int c4  = idx & 7;
      int gr  = rowBase + r;
      if (gr >= M) gr = M - 1;   // clamp (stores are guarded)
      v4f v = __builtin_nontemporal_load(
          (const v4f*)(A + (size_t)gr * K + kc + c4 * 4));
      *(v4f*)(&As[r][c4 * 4]) = v;
    }
    // Stage B chunk: TK x BN floats, pair-interleaved: (k,n) -> BsP[k/2][n][k&1]
    constexpr int BTOT4 = TK * BN / 4;
#pragma unroll
    for (int i = 0; i < BTOT4 / 256; ++i) {
      int idx = t + i * 256;
      int r   = idx / (BN / 4);
      int c4  = idx % (BN / 4);
      v4f v = *(const v4f*)(B + (size_t)(kc + r) * BN + c4 * 4);
      int p = r >> 1, par = r & 1, n = c4 * 4;
      BsP[p][n + 0][par] = v.x;
      BsP[p][n + 1][par] = v.y;
      BsP[p][n + 2][par] = v.z;
      BsP[p][n + 3][par] = v.w;
    }
    __syncthreads();

#pragma unroll
    for (int s = 0; s < TK / 4; ++s) {
      const int k = s * 4 + 2 * half;   // this lane-half's K pair base
      const int p = (s << 1) + half;    // pair-row index
      // A 16x4 frag: row l, K{k,k+1} contiguous -> one b64 per lane
      v2f a = *(const v2f*)(&As[m0w + l][k]);
#pragma unroll
      for (int nt = 0; nt < NT; ++nt) {
        const int n0 = n0w + nt * 16;
        // B 4x16 frag: (B[k][n0+l], B[k+1][n0+l]) contiguous -> one b64
        v2f b = *(const v2f*)(&BsP[p][n0 + l][0]);
        acc[nt] = __builtin_amdgcn_wmma_f32_16x16x4_f32(
            false, a, false, b, (short)0, acc[nt], false, false);
      }
    }
    __syncthreads();
  }

  // C/D layout: VGPR i holds rows i (lanes 0-15) and i+8 (lanes 16-31), N = l
#pragma unroll
  for (int nt = 0; nt < NT; ++nt) {
    const int col = n0w + nt * 16 + l;
#pragma unroll
    for (int i = 0; i < 8; ++i) {
      int row = rowBase + m0w + i + 8 * half;
      if (row < M) C[(size_t)row * BN + col] = acc[nt][i];
    }
  }
}

// ---------------------------------------------------------------------------
// Aggregation: self-loop init, edge scatter-add (atomics into L2-resident buf)
// ---------------------------------------------------------------------------
template <int C>
__global__ void self_loop_init_kernel(const float* __restrict__ h,
                                      const float* __restrict__ dinv,
                                      float* __restrict__ out, int n) {
  int idx = blockIdx.x * blockDim.x + threadIdx.x;
  if (idx < n * C) {
    int node = idx / C;
    float di = dinv[node];
    out[idx] = h[idx] * di * di;
  }
}

// 128 channels: wave per edge, one float4 per lane (32*4 = 128)
__global__ void scatter_edges_128_kernel(const long long* __restrict__ src,
                                         const long long* __restrict__ dst,
                                         const float* __restrict__ dinv,
                                         const float* __restrict__ h,
                                         float* __restrict__ out, int E) {
  int gid  = blockIdx.x * blockDim.x + threadIdx.x;
  int lane = gid & 31;
  int wid  = gid >> 5;
  int nw   = (gridDim.x * blockDim.x) >> 5;
  for (int e = wid; e < E; e += nw) {
    int s = (int)src[e];
    int d = (int)dst[e];
    float w = dinv[s] * dinv[d];
    v4f v = *(const v4f*)(h + (size_t)s * DIM_H + lane * 4);
    float* o = out + (size_t)d * DIM_H + lane * 4;
    atomicAdd(o + 0, v.x * w);
    atomicAdd(o + 1, v.y * w);
    atomicAdd(o + 2, v.z * w);
    atomicAdd(o + 3, v.w * w);
  }
}

// 64 channels: wave per edge, one float2 per lane (32*2 = 64)
__global__ void scatter_edges_64_kernel(const long long* __restrict__ src,
                                        const long long* __restrict__ dst,
                                        const float* __restrict__ dinv,
                                        const float* __restrict__ h,
                                        float* __restrict__ out, int E) {
  int gid  = blockIdx.x * blockDim.x + threadIdx.x;
  int lane = gid & 31;
  int wid  = gid >> 5;
  int nw   = (gridDim.x * blockDim.x) >> 5;
  for (int e = wid; e < E; e += nw) {
    int s = (int)src[e];
    int d = (int)dst[e];
    float w = dinv[s] * dinv[d];
    v2f v = *(const v2f*)(h + (size_t)s * DIM_OUT + lane * 2);
    float* o = out + (size_t)d * DIM_OUT + lane * 2;
    atomicAdd(o + 0, v.x * w);
    atomicAdd(o + 1, v.y * w);
  }
}

__global__ void bias_relu_kernel(float* __restrict__ g, const float* __restrict__ b,
                                 int n) {
  int idx = blockIdx.x * blockDim.x + threadIdx.x;
  if (idx < n * DIM_H) {
    int c = idx & (DIM_H - 1);
    g[idx] = fmaxf(g[idx] + b[c], 0.0f);
  }
}

// Fused bias + log_softmax over 64 cols: one wave32 per row, 2 cols per lane
__global__ void bias_logsoftmax_kernel(float* __restrict__ out,
                                       const float* __restrict__ b, int n) {
  int gid  = blockIdx.x * blockDim.x + threadIdx.x;
  int node = gid >> 5;
  int lane = gid & 31;
  if (node >= n) return;
  float* row = out + (size_t)node * DIM_OUT;
  float v0 = row[lane] + b[lane];
  float v1 = row[lane + 32] + b[lane + 32];
  float m = fmaxf(v0, v1);
#pragma unroll
  for (int off = 16; off > 0; off >>= 1) m = fmaxf(m, __shfl_xor(m, off, 32));
  float s = expf(v0 - m) + expf(v1 - m);
#pragma unroll
  for (int off = 16; off > 0; off >>= 1) s += __shfl_xor(s, off, 32);
  float ls = logf(s);
  row[lane] = v0 - m - ls;
  row[lane + 32] = v1 - m - ls;
}

// ---------------------------------------------------------------------------
// Launch
// ---------------------------------------------------------------------------
extern "C" void kernel_launch(void* const* d_in, const int* in_sizes, int n_in,
                              void* d_out, int out_size, void* d_ws, size_t ws_size,
                              hipStream_t stream) {
  (void)n_in; (void)out_size; (void)ws_size;
  const float*     x   = (const float*)d_in[0];
  const long long* ei  = (const long long*)d_in[1];  // int64 edge_index [2, E]
  const float*     W1  = (const float*)d_in[2];
  const float*     b1  = (const float*)d_in[3];
  const float*     W2  = (const float*)d_in[4];
  const float*     b2  = (const float*)d_in[5];
  float*           out = (float*)d_out;

  const int N = in_sizes[0] / DIM_IN;
  const int E = in_sizes[1] / 2;
  const long long* src = ei;
  const long long* dst = ei + E;

  float* ws   = (float*)d_ws;
  float* dinv = ws;                              // N floats (deg -> rsqrt in place)
  float* h1   = ws + (size_t)((N + 255) & ~255); // N * 128
  float* g1   = h1 + (size_t)N * DIM_H;          // N * 128
  float* h2   = h1;                              // reuse h1 after g1 is built

  const int tb = 256;

  // Normalization: deg = in-degree + 1 (self loop); dinv = rsqrt(deg)
  deg_init_kernel<<<(N + tb - 1) / tb, tb, 0, stream>>>(dinv, N);
  deg_accum_kernel<<<2048, tb, 0, stream>>>(dst, dinv, E);
  deg_rsqrt_kernel<<<(N + tb - 1) / tb, tb, 0, stream>>>(dinv, N);

  // Layer 1: h1 = x @ W1 (fp32 WMMA), aggregate, bias + ReLU
  gemm_wmma_f32<DIM_IN, DIM_H, 4><<<(N + 63) / 64, tb, 0, stream>>>(x, W1, h1, N);
  self_loop_init_kernel<DIM_H>
      <<<((size_t)N * DIM_H + tb - 1) / tb, tb, 0, stream>>>(h1, dinv, g1, N);
  scatter_edges_128_kernel<<<4096, tb, 0, stream>>>(src, dst, dinv, h1, g1, E);
  bias_relu_kernel<<<((size_t)N * DIM_H + tb - 1) / tb, tb, 0, stream>>>(g1, b1, N);

  // Layer 2: h2 = g1 @ W2 (fp32 WMMA), aggregate into d_out, bias + log_softmax
  gemm_wmma_f32<DIM_H, DIM_OUT, 2><<<(N + 63) / 64, tb, 0, stream>>>(g1, W2, h2, N);
  self_loop_init_kernel<DIM_OUT>
      <<<((size_t)N * DIM_OUT + tb - 1) / tb, tb, 0, stream>>>(h2, dinv, out, N);
  scatter_edges_64_kernel<<<4096, tb, 0, stream>>>(src, dst, dinv, h2, out, E);
  bias_logsoftmax_kernel<<<((size_t)N * 32 + tb - 1) / tb, tb, 0, stream>>>(out, b2, N);
}